// SC_ASC_pw_14516989461173
// MI455X (gfx1250) — compile-verified
//
#include <hip/hip_runtime.h>
#include <hip/hip_bf16.h>

// SC-ASC unrolled sparse-coding forward, fused single pass.
// N_PIX=262144, N_BANDS=224, N_END=16, unrollings=15.

#define N_PIX   262144
#define NB      224
#define NE      16
#define XPAD    228   // LDS row stride for x tile (stride%64=36 -> conflict-free, 8B aligned)
#define APAD    18    // LDS row stride for alpha scratch (even -> b64 aligned, conflict-free)

typedef __attribute__((ext_vector_type(2))) float v2f;
typedef __attribute__((ext_vector_type(4))) float v4f;
typedef __attribute__((ext_vector_type(8))) float v8f;

// ---------------------------------------------------------------------------
// Prep (one tiny block):
//   G2 = I - eta*(D^T D) + gamma*J   [16x16]  (J = all-ones; folds the
//        gamma*sum(alpha) penalty term into the iteration matmul)
//   Dp = pair-interleaved D for phase-1 B fragments:
//        Dp[p*32 + 2n + h] = D[(2p+h)*16 + n],  p=0..111, n=0..15, h=0..1
// ---------------------------------------------------------------------------
__global__ void sc_asc_prep(const float* __restrict__ D,
                            const float* __restrict__ etaPtr,
                            const float* __restrict__ gammaPtr,
                            float* __restrict__ ws) {
    const int i = threadIdx.x;                 // 256 threads
    // ---- G2
    {
        int r = i >> 4, c = i & 15;
        float s = 0.0f;
        for (int b = 0; b < NB; ++b)
            s += D[b * NE + r] * D[b * NE + c];
        ws[i] = ((r == c) ? 1.0f : 0.0f) - etaPtr[0] * s + gammaPtr[0];
    }
    // ---- Dp (3584 floats at ws+256)
    float* Dp = ws + 256;
    for (int t = 0; t < 14; ++t) {
        int idx  = i + t * 256;                // 0..3583
        int p    = idx >> 5;
        int rem  = idx & 31;
        int n    = rem >> 1;
        int h    = rem & 1;
        Dp[idx] = D[(2 * p + h) * NE + n];
    }
}

// ---------------------------------------------------------------------------
// Main fused kernel: 4 waves per WG, 16 pixels per wave.
// ---------------------------------------------------------------------------
__launch_bounds__(128)
__global__ void sc_asc_main(const float* __restrict__ x,
                            const float* __restrict__ D,
                            const float* __restrict__ etaPtr,
                            const float* __restrict__ gammaPtr,
                            const int*   __restrict__ unrollPtr,
                            const float* __restrict__ ws,
                            float* __restrict__ recon,
                            float* __restrict__ alphaOut) {
    __shared__ float lds[4 * (16 * XPAD + 16 * APAD)];   // 62976 B

    const int wave    = threadIdx.x >> 5;
    const int lane    = threadIdx.x & 31;
    const int ln      = lane & 15;             // N / M sub-index within half-wave
    const int halfSel = lane >> 4;             // 0: lanes 0-15, 1: lanes 16-31
    const int kOff    = halfSel * 2;           // K split of A/B fragments per half-wave

    float* xs = lds + wave * (16 * XPAD + 16 * APAD);
    float* as = xs + 16 * XPAD;

    const float* G2 = ws;                      // [16x16]
    const float* Dp = ws + 256;                // pair-interleaved D

    const float eta    = etaPtr[0];
    const float gamma  = gammaPtr[0];
    const int   unroll = unrollPtr[0];

    const int tileBase = (blockIdx.x * 4 + wave) * 16;   // first pixel of this wave's tile

    // ---- Stage x tile: 16 consecutive rows == contiguous 14336 B in global.
    {
        const float* xg = x + (size_t)tileBase * NB;
        #pragma unroll
        for (int j = 0; j < 28; ++j) {
            int idx4 = j * 32 + lane;          // float4 index in tile, 0..895
            int row  = idx4 / 56;              // 56 float4 per 224-float row
            int col4 = idx4 % 56;
            v4f v = __builtin_nontemporal_load((const v4f*)(xg + (size_t)idx4 * 4));
            *(v4f*)(xs + row * XPAD + col4 * 4) = v;
        }
    }
    __syncthreads();

    // ---- Phase 1: c = x_tile @ D  (16x224 * 224x16), 56 f32 WMMAs.
    v8f c = {};
    for (int kk = 0; kk < NB; kk += 4) {
        v2f a = *(const v2f*)(xs + ln * XPAD + kk + kOff);               // A: M=ln
        v2f b = *(const v2f*)(Dp + ((kk >> 1) + halfSel) * 32 + 2 * ln); // B: coalesced b64
        c = __builtin_amdgcn_wmma_f32_16x16x4_f32(
                false, a, false, b, (short)0, c, false, false);
    }

    // g = eta * (x@D); alpha0 = relu(g + eta*gamma)
    v8f g, alpha;
    #pragma unroll
    for (int r = 0; r < 8; ++r) {
        g[r]     = eta * c[r];
        alpha[r] = fmaxf(g[r] + eta * gamma, 0.0f);
    }

    // Preload G2 B-fragments (constant across iterations): 4 K-slices.
    v2f gb[4];
    #pragma unroll
    for (int q = 0; q < 4; ++q) {
        int k = q * 4 + kOff;
        gb[q].x = G2[k       * NE + ln];
        gb[q].y = G2[(k + 1) * NE + ln];
    }
    const v2f bgam = {gamma, gamma};           // B = gamma*J for row-sum WMMA

    // ---- Phase 2: unrolled iterations.
    //   s     = gamma*(sum(alpha_old)-1)        (4 WMMAs vs gamma*J, C=-gamma)
    //   lamAcc += s                             (lamAcc tracks lam_{t-1}, lam_{-1}=0)
    //   c2    = g + alpha_old @ G2^T            (contains gamma*sum(alpha_old) term)
    //   alpha = relu(c2 + lamAcc - gamma)
    v8f lamAcc = {};
    v2f af[4];
    for (int it = 0; it < unroll - 1; ++it) {
        // alpha (C layout) -> LDS [m][j], re-read as A fragments.
        __syncthreads();
        #pragma unroll
        for (int r = 0; r < 8; ++r)
            as[(r + halfSel * 8) * APAD + ln] = alpha[r];
        __syncthreads();
        #pragma unroll
        for (int q = 0; q < 4; ++q)
            af[q] = *(const v2f*)(as + ln * APAD + q * 4 + kOff);

        // row-sum term via matrix pipe (replaces cross-lane shuffle reduction)
        v8f s = {-gamma, -gamma, -gamma, -gamma, -gamma, -gamma, -gamma, -gamma};
        #pragma unroll
        for (int q = 0; q < 4; ++q)
            s = __builtin_amdgcn_wmma_f32_16x16x4_f32(
                    false, af[q], false, bgam, (short)0, s, false, false);

        // main iteration matmul, accumulating from C = g
        v8f c2 = g;
        #pragma unroll
        for (int q = 0; q < 4; ++q)
            c2 = __builtin_amdgcn_wmma_f32_16x16x4_f32(
                    false, af[q], false, gb[q], (short)0, c2, false, false);

        #pragma unroll
        for (int r = 0; r < 8; ++r) {
            lamAcc[r] += s[r];
            alpha[r] = fmaxf(c2[r] + (lamAcc[r] - gamma), 0.0f);
        }
    }

    // Final alpha -> A fragments (for recon GEMM).
    __syncthreads();
    #pragma unroll
    for (int r = 0; r < 8; ++r)
        as[(r + halfSel * 8) * APAD + ln] = alpha[r];
    __syncthreads();
    #pragma unroll
    for (int q = 0; q < 4; ++q)
        af[q] = *(const v2f*)(as + ln * APAD + q * 4 + kOff);

    // ---- Phase 3: recon = alpha @ D^T, 14 band tiles x 4 WMMAs.
    for (int t = 0; t < 14; ++t) {
        const int band = t * 16 + ln;
        v8f cr = {};
        #pragma unroll
        for (int q = 0; q < 4; ++q) {
            int k = q * 4 + kOff;
            v2f b = *(const v2f*)(D + band * NE + k);   // B[k][band] = D[band][k], b64
            cr = __builtin_amdgcn_wmma_f32_16x16x4_f32(
                    false, af[q], false, b, (short)0, cr, false, false);
        }
        #pragma unroll
        for (int r = 0; r < 8; ++r) {
            int m = tileBase + r + halfSel * 8;
            __builtin_nontemporal_store(cr[r], recon + (size_t)m * NB + band);
        }
    }

    // ---- alpha output [N,16]
    #pragma unroll
    for (int r = 0; r < 8; ++r) {
        int m = tileBase + r + halfSel * 8;
        __builtin_nontemporal_store(alpha[r], alphaOut + (size_t)m * NE + ln);
    }
}

// ---------------------------------------------------------------------------
extern "C" void kernel_launch(void* const* d_in, const int* in_sizes, int n_in,
                              void* d_out, int out_size, void* d_ws, size_t ws_size,
                              hipStream_t stream) {
    const float* x      = (const float*)d_in[0];
    const float* D      = (const float*)d_in[1];
    const float* eta    = (const float*)d_in[2];
    const float* gamma  = (const float*)d_in[3];
    const int*   unroll = (const int*)d_in[4];

    float* ws       = (float*)d_ws;                 // G2 (1 KB) + Dp (14 KB)
    float* recon    = (float*)d_out;                // [N, 224]
    float* alphaOut = recon + (size_t)N_PIX * NB;   // [N, 16]

    sc_asc_prep<<<1, 256, 0, stream>>>(D, eta, gamma, ws);
    sc_asc_main<<<N_PIX / 64, 128, 0, stream>>>(x, D, eta, gamma, unroll, ws,
                                                recon, alphaOut);
}